// PromptAttention_317827580706
// MI455X (gfx1250) — compile-verified
//
#include <hip/hip_runtime.h>
#include <cstdint>
#include <cstddef>

// Problem constants (match reference)
#define BN      4
#define SEQ_N   3144
#define CH      768
#define C3      2304          // 3*CH
#define NH      12
#define HD      64
#define NAUX    8             // 1 + NUM_PROMPT
#define NP      4             // NUM_PATCH
#define LSEQ    792           // NAUX + (SEQ_N-NAUX)/NP
#define KTILES  50            // ceil(LSEQ/16) -> pad to 800
#define NCHUNK  48            // CH/16 k-chunks

typedef uint32_t u32;
typedef __attribute__((ext_vector_type(2))) float v2f;
typedef __attribute__((ext_vector_type(8))) float v8f;
typedef __attribute__((ext_vector_type(4))) uint32_t v4u32;
typedef __attribute__((ext_vector_type(8))) int v8i32;
typedef __attribute__((ext_vector_type(4))) int v4i32;

__device__ __forceinline__ v8f wmma_f32(v2f a, v2f b, v8f c) {
  // D = A(16x4) * B(4x16) + C(16x16), fp32 WMMA on gfx1250
  return __builtin_amdgcn_wmma_f32_16x16x4_f32(false, a, false, b, (short)0, c,
                                               false, false);
}

// ---------------------------------------------------------------------------
// Tensor Data Mover: DMA a 16(row) x 64(col) fp32 tile (row stride
// `row_stride_elems`) from global memory into LDS at byte offset `lds_off`.
// D# layout per CDNA5 ISA 8.3/8.4; 2-D tensor -> groups 2/3 zero.
// 6-arg builtin form (clang-23 / therock-10.0 headers).
// ---------------------------------------------------------------------------
__device__ __forceinline__ void tdm_load_tile_16x64(u32 lds_off,
                                                    const void* gsrc,
                                                    u32 row_stride_elems) {
  uint64_t ga = (uint64_t)(uintptr_t)gsrc;
  v4u32 g0;
  g0[0] = 1u;                                           // count=1 (valid user D#)
  g0[1] = lds_off;                                      // lds_addr (bytes)
  g0[2] = (u32)ga;                                      // global_addr[31:0]
  g0[3] = ((u32)(ga >> 32) & 0x01ffffffu) | (2u << 30); // global_addr[56:32], type=2
  v8i32 g1;
  g1[0] = (int)(2u << 16);          // workgroup_mask=0, data_size=2 (4 bytes)
  g1[1] = (int)(64u << 16);         // tensor_dim0 = 64
  g1[2] = (int)(16u << 16);         // tensor_dim1 = 16
  g1[3] = (int)(64u << 16);         // tile_dim0 = 64
  g1[4] = (int)16u;                 // tile_dim1 = 16, tile_dim2 = 0
  g1[5] = (int)row_stride_elems;    // tensor_dim0_stride[31:0]
  g1[6] = 0;                        // stride hi, tensor_dim1_stride = 0
  g1[7] = 0;
  v4i32 z4; z4[0] = 0; z4[1] = 0; z4[2] = 0; z4[3] = 0;
  v8i32 z8;
#pragma unroll
  for (int i = 0; i < 8; ++i) z8[i] = 0;
  __builtin_amdgcn_tensor_load_to_lds(g0, g1, z4, z4, z8, 0);
}

// ---------------------------------------------------------------------------
// Kernel 1: qkv = x @ Wqkv      (M=12576, K=768, N=2304)
// Block = 3 waves x 32 rows = 96 rows, one shared 64-col group.
// W tiles (16x64) staged into LDS by TDM, double-buffered.
// ---------------------------------------------------------------------------
__global__ __launch_bounds__(96) void qkv_gemm(const float* __restrict__ X,
                                               const float* __restrict__ W,
                                               float* __restrict__ QKV) {
  __shared__ float ldsW[2][16 * 64];
  const int lane = threadIdx.x & 31;
  const int wid  = threadIdx.x >> 5;
  const int row0 = blockIdx.x * 96 + wid * 32;
  const int col0 = blockIdx.y * 64;
  const int m  = lane & 15;
  const int kb = (lane >> 4) << 1;

  v8f acc[2][4];
#pragma unroll
  for (int h = 0; h < 2; ++h)
#pragma unroll
    for (int t = 0; t < 4; ++t)
#pragma unroll
      for (int r = 0; r < 8; ++r) acc[h][t][r] = 0.0f;

  const float* xr0 = X + (size_t)(row0 + m) * CH + kb;
  const float* xr1 = X + (size_t)(row0 + 16 + m) * CH + kb;
  const bool issuer = (wid == 0);

  if (issuer)
    tdm_load_tile_16x64((u32)(uintptr_t)&ldsW[0][0], W + col0, C3);

  for (int c = 0; c < NCHUNK; ++c) {
    if (issuer) {
      if (c + 1 < NCHUNK) {
        tdm_load_tile_16x64((u32)(uintptr_t)&ldsW[(c + 1) & 1][0],
                            W + (size_t)(c + 1) * 16 * C3 + col0, C3);
        __builtin_amdgcn_s_wait_tensorcnt(1);   // chunk c resident (in-order)
      } else {
        __builtin_amdgcn_s_wait_tensorcnt(0);
      }
    }
    __syncthreads();
    const float* lb = &ldsW[c & 1][0];
    const int kg = c * 16;
#pragma unroll
    for (int s = 0; s < 4; ++s) {
      v2f a0; a0[0] = xr0[kg + s * 4]; a0[1] = xr0[kg + s * 4 + 1];
      v2f a1; a1[0] = xr1[kg + s * 4]; a1[1] = xr1[kg + s * 4 + 1];
      const int kl = s * 4 + kb;
#pragma unroll
      for (int t = 0; t < 4; ++t) {
        v2f b; b[0] = lb[kl * 64 + t * 16 + m];
        b[1] = lb[(kl + 1) * 64 + t * 16 + m];
        acc[0][t] = wmma_f32(a0, b, acc[0][t]);
        acc[1][t] = wmma_f32(a1, b, acc[1][t]);
      }
    }
    __syncthreads();   // protect buffer c&1 before next TDM overwrite
  }

  const int rb = (lane >> 4) * 8;
#pragma unroll
  for (int h = 0; h < 2; ++h)
#pragma unroll
    for (int t = 0; t < 4; ++t)
#pragma unroll
      for (int r = 0; r < 8; ++r)
        QKV[(size_t)(row0 + h * 16 + rb + r) * C3 + col0 + t * 16 + m] =
            acc[h][t][r];
}

// ---------------------------------------------------------------------------
// Kernel 2: flash attention per (b,p,h), one wave per 16-query tile.
// S = Q K^T (16 WMMAs) -> online softmax (shfl row reductions) ->
// P transposed via per-wave LDS tile -> O += P V (16 WMMAs).
// ---------------------------------------------------------------------------
__global__ __launch_bounds__(128) void attn(const float* __restrict__ QKV,
                                            float* __restrict__ Ypre,
                                            float* __restrict__ Aux) {
  __shared__ float ldsP[4][16 * 16];
  const int lane = threadIdx.x & 31;
  const int wid  = threadIdx.x >> 5;

  int task = blockIdx.x * 4 + wid;                  // 0..9599
  const int qtile = task % KTILES; task /= KTILES;  // 0..49
  const int h = task % NH;  task /= NH;
  const int p = task % NP;  task /= NP;
  const int b = task;                               // 0..3

  const int m  = lane & 15;
  const int kb = (lane >> 4) << 1;
  const int rb = (lane >> 4) * 8;
  const int coff = h * HD;
  const float scale = 0.125f;                       // 64^-0.5
  const float NEG = -3.0e38f;

  auto tok = [&](int j) -> int { return (j < NAUX) ? j : NAUX + (j - NAUX) * NP + p; };

  // Q fragments (A operand): 16 d-chunks of 4
  v2f aq[16];
  {
    int j = qtile * 16 + m;
    int t = tok(j < LSEQ ? j : 0);
    const float* qp = QKV + ((size_t)(b * SEQ_N + t) * 3 + 0) * CH + coff + kb;
#pragma unroll
    for (int c = 0; c < 16; ++c) {
      v2f a; a[0] = qp[c * 4] * scale; a[1] = qp[c * 4 + 1] * scale;
      aq[c] = a;
    }
  }

  v8f o[4];
  float mst[8], lst[8];
#pragma unroll
  for (int t = 0; t < 4; ++t)
#pragma unroll
    for (int r = 0; r < 8; ++r) o[t][r] = 0.0f;
#pragma unroll
  for (int r = 0; r < 8; ++r) { mst[r] = NEG; lst[r] = 0.0f; }

  float* lp = ldsP[wid];

  for (int kt = 0; kt < KTILES; ++kt) {
    // ---- S tile: cols = keys kt*16 + (lane&15)
    v8f s;
#pragma unroll
    for (int r = 0; r < 8; ++r) s[r] = 0.0f;
    const int jk = kt * 16 + m;
    const int tk = tok(jk < LSEQ ? jk : 0);
    const float* kp = QKV + ((size_t)(b * SEQ_N + tk) * 3 + 1) * CH + coff + kb;
#pragma unroll
    for (int c = 0; c < 16; ++c) {
      v2f bk; bk[0] = kp[c * 4]; bk[1] = kp[c * 4 + 1];
      s = wmma_f32(aq[c], bk, s);
    }
    if (jk >= LSEQ) {
#pragma unroll
      for (int r = 0; r < 8; ++r) s[r] = NEG;
    }

    // ---- online softmax over this key tile (rows live in 16-lane halves)
    float mx[8];
#pragma unroll
    for (int r = 0; r < 8; ++r) mx[r] = s[r];
#pragma unroll
    for (int off = 1; off < 16; off <<= 1)
#pragma unroll
      for (int r = 0; r < 8; ++r)
        mx[r] = fmaxf(mx[r], __shfl_xor(mx[r], off, 32));

    float pr[8], alpha[8];
#pragma unroll
    for (int r = 0; r < 8; ++r) {
      float mnew = fmaxf(mst[r], mx[r]);
      alpha[r] = __expf(mst[r] - mnew);
      pr[r]    = __expf(s[r] - mnew);
      mst[r]   = mnew;
    }
    float sm[8];
#pragma unroll
    for (int r = 0; r < 8; ++r) sm[r] = pr[r];
#pragma unroll
    for (int off = 1; off < 16; off <<= 1)
#pragma unroll
      for (int r = 0; r < 8; ++r) sm[r] += __shfl_xor(sm[r], off, 32);
#pragma unroll
    for (int r = 0; r < 8; ++r) lst[r] = lst[r] * alpha[r] + sm[r];
#pragma unroll
    for (int t = 0; t < 4; ++t)
#pragma unroll
      for (int r = 0; r < 8; ++r) o[t][r] *= alpha[r];

    // ---- transpose P (D layout -> A layout) through per-wave LDS tile
#pragma unroll
    for (int r = 0; r < 8; ++r) lp[(rb + r) * 16 + m] = pr[r];
    v2f ap[4];
#pragma unroll
    for (int kk = 0; kk < 4; ++kk) {
      v2f a; a[0] = lp[m * 16 + kk * 4 + kb]; a[1] = lp[m * 16 + kk * 4 + kb + 1];
      ap[kk] = a;
    }

    // ---- O += P * V
#pragma unroll
    for (int kk = 0; kk < 4; ++kk) {
      const int ja = kt * 16 + kk * 4 + kb;
      const int jb = ja + 1;
      const float* va =
          QKV + ((size_t)(b * SEQ_N + tok(ja < LSEQ ? ja : 0)) * 3 + 2) * CH + coff;
      const float* vb =
          QKV + ((size_t)(b * SEQ_N + tok(jb < LSEQ ? jb : 0)) * 3 + 2) * CH + coff;
#pragma unroll
      for (int t = 0; t < 4; ++t) {
        v2f bv; bv[0] = va[t * 16 + m]; bv[1] = vb[t * 16 + m];
        o[t] = wmma_f32(ap[kk], bv, o[t]);
      }
    }
  }

  // ---- finalize and scatter (aux tokens to per-p buffer, img tokens direct)
  float inv[8];
#pragma unroll
  for (int r = 0; r < 8; ++r) inv[r] = 1.0f / lst[r];
#pragma unroll
  for (int t = 0; t < 4; ++t)
#pragma unroll
    for (int r = 0; r < 8; ++r) {
      int j = qtile * 16 + rb + r;
      if (j >= LSEQ) continue;
      float val = o[t][r] * inv[r];
      int d = t * 16 + m;
      if (j < NAUX) {
        Aux[(((size_t)b * NP + p) * NAUX + j) * CH + coff + d] = val;
      } else {
        int n = NAUX + (j - NAUX) * NP + p;
        Ypre[((size_t)b * SEQ_N + n) * CH + coff + d] = val;
      }
    }
}

// ---------------------------------------------------------------------------
// Kernel 3: out = Yin @ Wproj + bproj, where Yin rows n<8 average the 4 aux
// copies. Same TDM-staged WMMA tiling as kernel 1 (N=768).
// ---------------------------------------------------------------------------
struct RowSrc { const float* p0; const float* p1; const float* p2; const float* p3; int aux; };

__global__ __launch_bounds__(96) void proj_gemm(const float* __restrict__ Ypre,
                                                const float* __restrict__ Aux,
                                                const float* __restrict__ W,
                                                const float* __restrict__ bias,
                                                float* __restrict__ Out) {
  __shared__ float ldsW[2][16 * 64];
  const int lane = threadIdx.x & 31;
  const int wid  = threadIdx.x >> 5;
  const int row0 = blockIdx.x * 96 + wid * 32;
  const int col0 = blockIdx.y * 64;                 // 0..11 groups
  const int m  = lane & 15;
  const int kb = (lane >> 4) << 1;

  auto mkrow = [&](int rg) -> RowSrc {
    int bb = rg / SEQ_N, nn = rg % SEQ_N;
    RowSrc rs;
    if (nn < NAUX) {
      rs.aux = 1;
      rs.p0 = Aux + (((size_t)bb * NP + 0) * NAUX + nn) * CH + kb;
      rs.p1 = Aux + (((size_t)bb * NP + 1) * NAUX + nn) * CH + kb;
      rs.p2 = Aux + (((size_t)bb * NP + 2) * NAUX + nn) * CH + kb;
      rs.p3 = Aux + (((size_t)bb * NP + 3) * NAUX + nn) * CH + kb;
    } else {
      rs.aux = 0;
      rs.p0 = Ypre + ((size_t)bb * SEQ_N + nn) * CH + kb;
      rs.p1 = rs.p0; rs.p2 = rs.p0; rs.p3 = rs.p0;
    }
    return rs;
  };
  RowSrc r0 = mkrow(row0 + m);
  RowSrc r1 = mkrow(row0 + 16 + m);

  auto lda = [&](const RowSrc& rs, int k0) -> v2f {
    v2f a;
    if (rs.aux) {
      a[0] = 0.25f * (rs.p0[k0] + rs.p1[k0] + rs.p2[k0] + rs.p3[k0]);
      a[1] = 0.25f * (rs.p0[k0 + 1] + rs.p1[k0 + 1] + rs.p2[k0 + 1] + rs.p3[k0 + 1]);
    } else {
      a[0] = rs.p0[k0];
      a[1] = rs.p0[k0 + 1];
    }
    return a;
  };

  v8f acc[2][4];
#pragma unroll
  for (int h = 0; h < 2; ++h)
#pragma unroll
    for (int t = 0; t < 4; ++t)
#pragma unroll
      for (int r = 0; r < 8; ++r) acc[h][t][r] = 0.0f;

  const bool issuer = (wid == 0);
  if (issuer)
    tdm_load_tile_16x64((u32)(uintptr_t)&ldsW[0][0], W + col0, CH);

  for (int c = 0; c < NCHUNK; ++c) {
    if (issuer) {
      if (c + 1 < NCHUNK) {
        tdm_load_tile_16x64((u32)(uintptr_t)&ldsW[(c + 1) & 1][0],
                            W + (size_t)(c + 1) * 16 * CH + col0, CH);
        __builtin_amdgcn_s_wait_tensorcnt(1);
      } else {
        __builtin_amdgcn_s_wait_tensorcnt(0);
      }
    }
    __syncthreads();
    const float* lb = &ldsW[c & 1][0];
    const int kg = c * 16;
#pragma unroll
    for (int s = 0; s < 4; ++s) {
      v2f a0 = lda(r0, kg + s * 4);
      v2f a1 = lda(r1, kg + s * 4);
      const int kl = s * 4 + kb;
#pragma unroll
      for (int t = 0; t < 4; ++t) {
        v2f b; b[0] = lb[kl * 64 + t * 16 + m];
        b[1] = lb[(kl + 1) * 64 + t * 16 + m];
        acc[0][t] = wmma_f32(a0, b, acc[0][t]);
        acc[1][t] = wmma_f32(a1, b, acc[1][t]);
      }
    }
    __syncthreads();
  }

  const int rb = (lane >> 4) * 8;
#pragma unroll
  for (int h = 0; h < 2; ++h)
#pragma unroll
    for (int t = 0; t < 4; ++t) {
      float bz = bias[col0 + t * 16 + m];
#pragma unroll
      for (int r = 0; r < 8; ++r)
        Out[(size_t)(row0 + h * 16 + rb + r) * CH + col0 + t * 16 + m] =
            acc[h][t][r] + bz;
    }
}

// ---------------------------------------------------------------------------
extern "C" void kernel_launch(void* const* d_in, const int* in_sizes, int n_in,
                              void* d_out, int out_size, void* d_ws, size_t ws_size,
                              hipStream_t stream) {
  const float* x     = (const float*)d_in[0];
  const float* Wqkv  = (const float*)d_in[1];
  const float* Wproj = (const float*)d_in[2];
  const float* bproj = (const float*)d_in[3];
  float* out = (float*)d_out;

  float* qkv  = (float*)d_ws;                              // B*N*3C
  float* ypre = qkv + (size_t)BN * SEQ_N * C3;             // B*N*C (img rows)
  float* aux  = ypre + (size_t)BN * SEQ_N * CH;            // B*P*NAUX*C

  // 1) QKV GEMM: 131 row-blocks of 96, 36 col groups of 64 (TDM-staged W)
  qkv_gemm<<<dim3(131, 36), 96, 0, stream>>>(x, Wqkv, qkv);

  // 2) attention: 4*4*12*50 = 9600 wave-tasks, 4 waves/block
  attn<<<dim3(2400), 128, 0, stream>>>(qkv, ypre, aux);

  // 3) projection: 131 row-blocks of 96, 12 col groups of 64 (TDM-staged W)
  proj_gemm<<<dim3(131, 12), 96, 0, stream>>>(ypre, aux, Wproj, bproj, out);
}